// TestAllReduceFusedAddRMSNormModel_69784628625431
// MI455X (gfx1250) — compile-verified
//
#include <hip/hip_runtime.h>
#include <hip/hip_bf16.h>

// RMSNorm-fused kernel for MI455X (gfx1250, wave32).
// x = 2*relu(h); var = mean(x^2) over hidden dim; norm = x*rsqrt(var+eps)*w.
// Outputs: d_out[0 .. R*H)   = norm
//          d_out[R*H .. 2RH) = x
//
// Bandwidth-bound (384 MiB @ 23.3 TB/s ~= 17 us). Row is register-resident
// (read HBM once), streamed with non-temporal b128 loads/stores. The 32-lane
// sum-of-squares all-reduce(+broadcast) is done with a chain of 5
// V_WMMA_F32_16X16X4_F32 ops (B/A = ones trick), cross-wave via LDS.

typedef __attribute__((ext_vector_type(2))) float v2f;
typedef __attribute__((ext_vector_type(4))) float v4f;
typedef __attribute__((ext_vector_type(8))) float v8f;

#define HID 4096
#define BLK 256          // 8 wave32s
#define V4_PER_THREAD 4  // 4096 floats / 256 threads / 4 per float4

// Full 32-lane sum with result broadcast to every lane, using WMMA f32 16x16x4.
// Step 1: A[m][0]=P[m] (lanes 0-15), A[m][2]=P[m] (lanes 16-31), other K = 0,
//         B = ones  ->  D[m][n] = P[m] + P[m+16] = S_m, replicated over n.
//         d0..d7 hold S_0..S_15 laid out exactly as a B-matrix expects.
// Step 2: A = ones, B = (d_{2i}, d_{2i+1}), accumulate over i=0..3:
//         each WMMA adds sum_k B[k][n] = 4 of the S_m; final acc = sum of all
//         32 partials in every element of D -> every lane.
__device__ __forceinline__ float wave_allreduce_sum_wmma(float partial) {
    v2f a;    a[0] = partial; a[1] = 0.0f;
    v2f ones; ones[0] = 1.0f; ones[1] = 1.0f;
    v8f zero = {};
    v8f d = __builtin_amdgcn_wmma_f32_16x16x4_f32(
        false, a, false, ones, (short)0, zero, false, false);
    v8f acc = {};
    v2f b;
    b[0] = d[0]; b[1] = d[1];
    acc = __builtin_amdgcn_wmma_f32_16x16x4_f32(
        false, ones, false, b, (short)0, acc, false, false);
    b[0] = d[2]; b[1] = d[3];
    acc = __builtin_amdgcn_wmma_f32_16x16x4_f32(
        false, ones, false, b, (short)0, acc, false, false);
    b[0] = d[4]; b[1] = d[5];
    acc = __builtin_amdgcn_wmma_f32_16x16x4_f32(
        false, ones, false, b, (short)0, acc, false, false);
    b[0] = d[6]; b[1] = d[7];
    acc = __builtin_amdgcn_wmma_f32_16x16x4_f32(
        false, ones, false, b, (short)0, acc, false, false);
    return acc[0];
}

__global__ void __launch_bounds__(BLK)
fused_relu_add_rmsnorm_kernel(const float* __restrict__ hs,
                              const float* __restrict__ weight,
                              float* __restrict__ norm_out,
                              float* __restrict__ x_out) {
    const int row = blockIdx.x;
    const int t   = threadIdx.x;

    const v4f* __restrict__ in4 = (const v4f*)(hs + (size_t)row * HID);
    v4f* __restrict__ x4  = (v4f*)(x_out    + (size_t)row * HID);
    v4f* __restrict__ n4  = (v4f*)(norm_out + (size_t)row * HID);
    const v4f* __restrict__ w4 = (const v4f*)weight;

    // Pass 1: stream row in (NT b128), compute x = 2*relu(h), write x (NT),
    // keep x register-resident, accumulate per-lane sum of squares.
    v4f xv[V4_PER_THREAD];
    float sumsq = 0.0f;
#pragma unroll
    for (int i = 0; i < V4_PER_THREAD; ++i) {
        const int idx = t + i * BLK;                 // coalesced b128
        v4f h = __builtin_nontemporal_load(&in4[idx]);
        v4f x;
        x[0] = 2.0f * fmaxf(h[0], 0.0f);
        x[1] = 2.0f * fmaxf(h[1], 0.0f);
        x[2] = 2.0f * fmaxf(h[2], 0.0f);
        x[3] = 2.0f * fmaxf(h[3], 0.0f);
        sumsq = fmaf(x[0], x[0], sumsq);
        sumsq = fmaf(x[1], x[1], sumsq);
        sumsq = fmaf(x[2], x[2], sumsq);
        sumsq = fmaf(x[3], x[3], sumsq);
        xv[i] = x;
        __builtin_nontemporal_store(x, &x4[idx]);
    }

    // 32-lane all-reduce via WMMA chain (EXEC is all-1s here: no divergence).
    float wave_sum = wave_allreduce_sum_wmma(sumsq);

    // Cross-wave combine (8 waves) through LDS.
    __shared__ float s_part[BLK / 32];
    const int wid  = t >> 5;
    const int lane = t & 31;
    if (lane == 0) s_part[wid] = wave_sum;
    __syncthreads();
    float total = 0.0f;
#pragma unroll
    for (int i = 0; i < BLK / 32; ++i) total += s_part[i];

    const float inv_rms = rsqrtf(total * (1.0f / (float)HID) + 1e-6f);

    // Pass 2: scale register-resident x by inv_rms * weight, stream out (NT).
    // weight is reused by all 8192 blocks -> regular (cached) load.
#pragma unroll
    for (int i = 0; i < V4_PER_THREAD; ++i) {
        const int idx = t + i * BLK;
        v4f w = w4[idx];
        v4f n;
        n[0] = xv[i][0] * inv_rms * w[0];
        n[1] = xv[i][1] * inv_rms * w[1];
        n[2] = xv[i][2] * inv_rms * w[2];
        n[3] = xv[i][3] * inv_rms * w[3];
        __builtin_nontemporal_store(n, &n4[idx]);
    }
}

extern "C" void kernel_launch(void* const* d_in, const int* in_sizes, int n_in,
                              void* d_out, int out_size, void* d_ws, size_t ws_size,
                              hipStream_t stream) {
    const float* hs = (const float*)d_in[0];     // (rows, 4096) fp32
    const float* w  = (const float*)d_in[1];     // (4096,) fp32
    const int rows  = in_sizes[0] / HID;

    float* norm_out = (float*)d_out;                      // first output
    float* x_out    = (float*)d_out + (size_t)rows * HID; // second output

    fused_relu_add_rmsnorm_kernel<<<rows, BLK, 0, stream>>>(hs, w, norm_out, x_out);
}